// Huber_25159918420340
// MI455X (gfx1250) — compile-verified
//
#include <hip/hip_runtime.h>
#include <math.h>

#define N1 3000
#define N2 3000
#define RANK 20
#define LAYERS 3
#define ITERS 3
#define MT ((N1 + 15) / 16)        // 188 row tiles
#define NT ((N2 + 15) / 16)        // 188 col tiles
#define NJ 4                       // col tiles per wave
#define NGX ((NT + NJ - 1) / NJ)   // 47

typedef __attribute__((ext_vector_type(2))) float v2f;
typedef __attribute__((ext_vector_type(8))) float v8f;

// ---------------------------------------------------------------- utilities
__global__ void hub_copy(const float* __restrict__ src, float* __restrict__ dst, int n) {
  int i = blockIdx.x * blockDim.x + threadIdx.x;
  if (i < n) dst[i] = src[i];
}

// vp == nullptr -> fill with 0
__global__ void hub_fill(float* __restrict__ dst, int n, const float* __restrict__ vp) {
  int i = blockIdx.x * blockDim.x + threadIdx.x;
  if (i < n) dst[i] = vp ? vp[0] : 0.0f;
}

// alpha = c*(1-erf(sqrt(c))) + 0.5*(erf(c/sqrt2) - c*sqrt(2/pi)*exp(-c^2/2))
// (closed forms of chi2_cdf(2c,1) and chi2_cdf(c^2,3))
__global__ void hub_alpha(const float* __restrict__ c_list, float* __restrict__ alpha) {
  if (blockIdx.x == 0 && threadIdx.x == 0) {
    for (int l = 0; l < LAYERS; l++) {
      float c = c_list[l];
      alpha[l] = c * (1.0f - erff(sqrtf(c)))
               + 0.5f * (erff(c * 0.70710678118f)
                         - c * 0.79788456080f * expf(-0.5f * c * c));
    }
  }
}

__global__ void hub_nobs(const float* __restrict__ X, float* __restrict__ nobs) {
  int j = blockIdx.x * blockDim.x + threadIdx.x;
  if (j < N2) {
    float cnt = 0.0f;
    for (int i = 0; i < N1; i++) cnt += (X[(size_t)i * N2 + j] != 0.0f) ? 1.0f : 0.0f;
    nobs[j] = cnt;
  }
}

// --------------------------------------------------- fused UV + Huber psi (WMMA f32)
// One wave32 per (16 x 64) strip: 4 column tiles, K=20 = 5 x V_WMMA_F32_16X16X4_F32
// per tile (20 WMMAs/wave), A fragments reused across tiles.
// All loads are branchless with clamped addresses: out-of-range A rows / B cols only
// pollute out-of-range output rows/cols, which are never stored or accumulated.
// Division by sigma is hoisted to one reciprocal per column tile (MODE 0) or per
// row (MODE 1) and applied as a multiply.
// MODE 0: column pass (sig per column, ssum += per-column sum Psi^2)
// MODE 1: row pass    (sig per row,    ssum += per-row    sum Psi^2)
// MODE 2: write U@V to outUV (final output)
template <int MODE>
__global__ __launch_bounds__(32) void hub_uv_psi(
    const float* __restrict__ X, const float* __restrict__ U,
    const float* __restrict__ V, const float* __restrict__ sig,
    const float* __restrict__ c_list, int layer,
    float* __restrict__ Psi, float* __restrict__ ssum,
    float* __restrict__ outUV)
{
  const int lane = threadIdx.x;
  const int half = lane >> 4;            // 0: lanes 0-15, 1: lanes 16-31
  const int lm   = lane & 15;
  const int i0 = blockIdx.y * 16;
  const int jg = blockIdx.x * (16 * NJ);
  const int arow = i0 + lm;              // A row (M index) for this lane
  const bool interior = (i0 + 16 <= N1) && (jg + 16 * NJ <= N2);  // wave-uniform

  // ---- A fragments: contiguous K pair -> one b64 load each, clamped address ----
  v2f a[5];
  {
    const int ar = min(arow, N1 - 1);
    const v2f* Urow = (const v2f*)(U + ar * RANK);
    #pragma unroll
    for (int kk = 0; kk < 5; kk++) a[kk] = Urow[2 * kk + half];
  }

  const float cval = (MODE == 2) ? 0.0f : c_list[layer];

  // row-pass per-row 1/sigma + Psi^2 partials, shared across the 4 tiles
  float sgi[8], rs[8];
  if (MODE == 1) {
    #pragma unroll
    for (int v = 0; v < 8; v++) {
      int row = i0 + v + 8 * half;
      sgi[v] = 1.0f / sig[min(row, N1 - 1)];
      rs[v] = 0.0f;
    }
  }

  #pragma unroll
  for (int t = 0; t < NJ; t++) {
    const int j0 = jg + 16 * t;
    const int col = j0 + lm;             // B col (N index) for this lane
    const int cc  = min(col, N2 - 1);    // clamped (see note above)

    // ---- B fragments (branchless) ----
    v2f b[5];
    #pragma unroll
    for (int kk = 0; kk < 5; kk++) {
      int ka = 4 * kk + 2 * half;
      b[kk].x = V[(size_t)ka       * N2 + cc];
      b[kk].y = V[(size_t)(ka + 1) * N2 + cc];
    }

    // ---- 16x16xK=20 product ----
    v8f acc = {0.0f, 0.0f, 0.0f, 0.0f, 0.0f, 0.0f, 0.0f, 0.0f};
    #pragma unroll
    for (int kk = 0; kk < 5; kk++)
      acc = __builtin_amdgcn_wmma_f32_16x16x4_f32(false, a[kk], false, b[kk],
                                                  (short)0, acc, false, false);

    // ---- epilogue ----
    if (MODE == 2) {
      if (interior) {
        #pragma unroll
        for (int v = 0; v < 8; v++) {
          int row = i0 + v + 8 * half;   // C layout: VGPR v, lane-half h -> M = v + 8h
          outUV[(size_t)row * N2 + col] = acc[v];
        }
      } else {
        #pragma unroll
        for (int v = 0; v < 8; v++) {
          int row = i0 + v + 8 * half;
          if (row < N1 && col < N2) outUV[(size_t)row * N2 + col] = acc[v];
        }
      }
    } else {
      const float sinv = (MODE == 0) ? (1.0f / sig[cc]) : 0.0f;  // one div per tile
      float psi_v[8];
      float csum = 0.0f;
      #pragma unroll
      for (int v = 0; v < 8; v++) {
        int row = i0 + v + 8 * half;
        bool inb = interior || (row < N1 && col < N2);
        float x = X[(size_t)min(row, N1 - 1) * N2 + cc];  // clamped, branchless
        float si = (MODE == 0) ? sinv : sgi[v];
        float tc = (x - acc[v]) * si;                     // R/sig on observed entries
        tc = fminf(fmaxf(tc, -cval), cval);               // Huber psi
        float psi = (inb && x != 0.0f) ? tc : 0.0f;       // mask + bounds (selects)
        psi_v[v] = psi;
        if (MODE == 0) csum += psi * psi; else rs[v] += psi * psi;
      }
      if (interior) {
        #pragma unroll
        for (int v = 0; v < 8; v++) {
          int row = i0 + v + 8 * half;
          Psi[(size_t)row * N2 + col] = psi_v[v];
        }
      } else {
        #pragma unroll
        for (int v = 0; v < 8; v++) {
          int row = i0 + v + 8 * half;
          if (row < N1 && col < N2) Psi[(size_t)row * N2 + col] = psi_v[v];
        }
      }
      if (MODE == 0) {
        if (col < N2) atomicAdd(&ssum[col], csum);
      }
    }
  }

  if (MODE == 1) {
    #pragma unroll
    for (int v = 0; v < 8; v++) {        // reduce over the 16 N-lanes of each half
      float x = rs[v];
      x += __shfl_xor(x, 8, 16);
      x += __shfl_xor(x, 4, 16);
      x += __shfl_xor(x, 2, 16);
      x += __shfl_xor(x, 1, 16);
      rs[v] = x;
    }
    if (lm == 0) {
      #pragma unroll
      for (int v = 0; v < 8; v++) {
        int row = i0 + v + 8 * half;
        if (row < N1) atomicAdd(&ssum[row], rs[v]);
      }
    }
  }
}

// ------------------------------------------------------------- masked Grams
#define GR_CH 64
// G[j] = U^T diag(mask[:,j]) U  -> (N2, 20, 20); one block per column
__global__ __launch_bounds__(512) void hub_gram_col(
    const float* __restrict__ X, const float* __restrict__ U, float* __restrict__ G)
{
  __shared__ float Ul[GR_CH * RANK];
  __shared__ float Ml[GR_CH];
  const int j = blockIdx.x;
  const int t = threadIdx.x;
  const int r = t / RANK, s = t % RANK;
  float acc = 0.0f;
  for (int c0 = 0; c0 < N1; c0 += GR_CH) {
    for (int e = t; e < GR_CH * RANK; e += 512) {
      int ii = e / RANK, rr = e % RANK;
      int row = c0 + ii;
      Ul[e] = (row < N1) ? U[row * RANK + rr] : 0.0f;
    }
    if (t < GR_CH) {
      int row = c0 + t;
      Ml[t] = (row < N1 && X[(size_t)row * N2 + j] != 0.0f) ? 1.0f : 0.0f;
    }
    __syncthreads();
    if (t < RANK * RANK) {
      #pragma unroll 8
      for (int ii = 0; ii < GR_CH; ii++)
        acc += Ml[ii] * Ul[ii * RANK + r] * Ul[ii * RANK + s];
    }
    __syncthreads();
  }
  if (t < RANK * RANK) G[(size_t)j * RANK * RANK + t] = acc;
}

// Gr[i] = V diag(mask[i,:]) V^T -> (N1, 20, 20); one block per row
__global__ __launch_bounds__(512) void hub_gram_row(
    const float* __restrict__ X, const float* __restrict__ V, float* __restrict__ G)
{
  __shared__ float Vl[RANK * GR_CH];
  __shared__ float Ml[GR_CH];
  const int i = blockIdx.x;
  const int t = threadIdx.x;
  const int r = t / RANK, s = t % RANK;
  float acc = 0.0f;
  for (int c0 = 0; c0 < N2; c0 += GR_CH) {
    for (int e = t; e < RANK * GR_CH; e += 512) {
      int rr = e / GR_CH, jj = e % GR_CH;
      int col = c0 + jj;
      Vl[e] = (col < N2) ? V[(size_t)rr * N2 + col] : 0.0f;
    }
    if (t < GR_CH) {
      int col = c0 + t;
      Ml[t] = (col < N2 && X[(size_t)i * N2 + col] != 0.0f) ? 1.0f : 0.0f;
    }
    __syncthreads();
    if (t < RANK * RANK) {
      #pragma unroll 8
      for (int jj = 0; jj < GR_CH; jj++)
        acc += Ml[jj] * Vl[r * GR_CH + jj] * Vl[s * GR_CH + jj];
    }
    __syncthreads();
  }
  if (t < RANK * RANK) G[(size_t)i * RANK * RANK + t] = acc;
}

// ---------------------------------------------- 20x20 inverse, Gauss-Jordan w/ pivot
__global__ __launch_bounds__(64) void hub_invert(
    const float* __restrict__ G, float* __restrict__ Gi)
{
  __shared__ float aug[RANK * 2 * RANK];   // 20 x 40 augmented [A | I]
  __shared__ float fcol[RANK];
  __shared__ int piv;
  const int m = blockIdx.x;
  const int t = threadIdx.x;
  const int W = 2 * RANK;
  for (int e = t; e < RANK * W; e += 64) {
    int r = e / W, c = e % W;
    aug[e] = (c < RANK) ? G[(size_t)m * RANK * RANK + r * RANK + c]
                        : ((c - RANK == r) ? 1.0f : 0.0f);
  }
  __syncthreads();
  for (int k = 0; k < RANK; k++) {
    if (t == 0) {
      int p = k; float best = fabsf(aug[k * W + k]);
      for (int i2 = k + 1; i2 < RANK; i2++) {
        float v = fabsf(aug[i2 * W + k]);
        if (v > best) { best = v; p = i2; }
      }
      piv = p;
    }
    __syncthreads();
    const int p = piv;
    if (p != k && t < W) {
      float tmp = aug[k * W + t]; aug[k * W + t] = aug[p * W + t]; aug[p * W + t] = tmp;
    }
    __syncthreads();
    const float pv = aug[k * W + k];
    __syncthreads();
    if (t < W) aug[k * W + t] /= pv;
    if (t < RANK) fcol[t] = (t == k) ? 0.0f : aug[t * W + k];
    __syncthreads();
    for (int e = t; e < RANK * W; e += 64) {
      int r = e / W, c = e % W;
      if (r != k) aug[e] -= fcol[r] * aug[k * W + c];
    }
    __syncthreads();
  }
  for (int e = t; e < RANK * RANK; e += 64) {
    int r = e / RANK, c = e % RANK;
    Gi[(size_t)m * RANK * RANK + e] = aug[r * W + RANK + c];
  }
}

// ----------------------------------------------------------- sigma update
__global__ void hub_sig(const float* __restrict__ ssum, const float* __restrict__ nobs,
                        const float* __restrict__ alpha, const float* __restrict__ lam_list,
                        int layer, float* __restrict__ sig, int n, int mode)
{
  int i = blockIdx.x * blockDim.x + threadIdx.x;
  if (i >= n) return;
  float a = alpha[layer];
  float lam = lam_list[layer];
  float denom = (mode == 0) ? sqrtf(2.0f * nobs[i] * a) : sqrtf(2.0f * a);
  sig[i] = lam * sqrtf(ssum[i]) / denom;
}

// ------------------------------------------ W = U^T Psi (split-K, atomic accumulate)
#define TC_COLS 64
#define TC_ROWCHUNK 300
__global__ __launch_bounds__(320) void hub_tcol(
    const float* __restrict__ Psi, const float* __restrict__ U, float* __restrict__ T)
{
  const int t = threadIdx.x;
  const int j = blockIdx.x * TC_COLS + (t & 63);
  const int rg = t >> 6;                         // 0..4, four r each
  const int ibeg = blockIdx.y * TC_ROWCHUNK;
  const int iend = min(ibeg + TC_ROWCHUNK, N1);
  float a0 = 0.0f, a1 = 0.0f, a2 = 0.0f, a3 = 0.0f;
  for (int i = ibeg; i < iend; i++) {
    float p = (j < N2) ? Psi[(size_t)i * N2 + j] : 0.0f;
    const float* Ur = &U[i * RANK + rg * 4];
    a0 += Ur[0] * p; a1 += Ur[1] * p; a2 += Ur[2] * p; a3 += Ur[3] * p;
  }
  if (j < N2) {
    atomicAdd(&T[(size_t)(rg * 4 + 0) * N2 + j], a0);
    atomicAdd(&T[(size_t)(rg * 4 + 1) * N2 + j], a1);
    atomicAdd(&T[(size_t)(rg * 4 + 2) * N2 + j], a2);
    atomicAdd(&T[(size_t)(rg * 4 + 3) * N2 + j], a3);
  }
}

// V[r,j] += mu * sig[j] * sum_s Ginv[j,r,s] * W[s,j]
__global__ void hub_vupdate(float* __restrict__ V, const float* __restrict__ Gi,
                            const float* __restrict__ T, const float* __restrict__ sig,
                            const float* __restrict__ mu_list, int layer)
{
  int idx = blockIdx.x * blockDim.x + threadIdx.x;
  if (idx >= RANK * N2) return;
  int r = idx / N2, j = idx % N2;
  float acc = 0.0f;
  const float* g = &Gi[(size_t)j * RANK * RANK + r * RANK];
  #pragma unroll
  for (int s = 0; s < RANK; s++) acc += g[s] * T[(size_t)s * N2 + j];
  V[idx] += mu_list[layer] * sig[j] * acc;
}

// ------------------------------------------ T = Psi V^T, one block per row
__global__ __launch_bounds__(256) void hub_trow(
    const float* __restrict__ Psi, const float* __restrict__ V, float* __restrict__ T)
{
  const int i = blockIdx.x;
  const int t = threadIdx.x;
  float acc[RANK];
  #pragma unroll
  for (int r = 0; r < RANK; r++) acc[r] = 0.0f;
  for (int j = t; j < N2; j += 256) {
    float p = Psi[(size_t)i * N2 + j];
    #pragma unroll
    for (int r = 0; r < RANK; r++) acc[r] += p * V[(size_t)r * N2 + j];
  }
  __shared__ float sred[RANK];
  if (t < RANK) sred[t] = 0.0f;
  __syncthreads();
  #pragma unroll
  for (int r = 0; r < RANK; r++) {
    float x = acc[r];
    x += __shfl_down(x, 16, 32);
    x += __shfl_down(x, 8, 32);
    x += __shfl_down(x, 4, 32);
    x += __shfl_down(x, 2, 32);
    x += __shfl_down(x, 1, 32);
    acc[r] = x;
  }
  if ((t & 31) == 0)
    for (int r = 0; r < RANK; r++) atomicAdd(&sred[r], acc[r]);
  __syncthreads();
  if (t < RANK) T[(size_t)i * RANK + t] = sred[t];
}

// U[i,r] += mu * sig[i] * sum_s Grinv[i,r,s] * T[i,s]
__global__ void hub_uupdate(float* __restrict__ U, const float* __restrict__ Gi,
                            const float* __restrict__ T, const float* __restrict__ sig,
                            const float* __restrict__ mu_list, int layer)
{
  int idx = blockIdx.x * blockDim.x + threadIdx.x;
  if (idx >= N1 * RANK) return;
  int i = idx / RANK, r = idx % RANK;
  float acc = 0.0f;
  const float* g  = &Gi[(size_t)i * RANK * RANK + r * RANK];
  const float* tv = &T[(size_t)i * RANK];
  #pragma unroll
  for (int s = 0; s < RANK; s++) acc += g[s] * tv[s];
  U[idx] += mu_list[layer] * sig[i] * acc;
}

// --------------------------------------------------------------- launcher
extern "C" void kernel_launch(void* const* d_in, const int* in_sizes, int n_in,
                              void* d_out, int out_size, void* d_ws, size_t ws_size,
                              hipStream_t stream)
{
  const float* X      = (const float*)d_in[0];
  const float* U0     = (const float*)d_in[1];
  const float* V0     = (const float*)d_in[2];
  const float* sigma0 = (const float*)d_in[3];
  const float* c_list = (const float*)d_in[4];
  const float* lamda  = (const float*)d_in[5];
  const float* mu     = (const float*)d_in[6];
  (void)in_sizes; (void)n_in; (void)out_size; (void)ws_size;

  // workspace carve-up (~46.3 MB of f32)
  float* ws    = (float*)d_ws;
  float* Psi   = ws;  ws += (size_t)N1 * N2;          // 9,000,000
  float* G     = ws;  ws += (size_t)N2 * RANK * RANK; // 1,200,000
  float* Gi    = ws;  ws += (size_t)N2 * RANK * RANK; // 1,200,000
  float* Ucur  = ws;  ws += (size_t)N1 * RANK;        //    60,000
  float* Vcur  = ws;  ws += (size_t)RANK * N2;        //    60,000
  float* T     = ws;  ws += (size_t)RANK * N2;        //    60,000 (also N1*RANK)
  float* ssum  = ws;  ws += N1;                       //     3,000
  float* sig   = ws;  ws += N1;                       //     3,000
  float* nobs  = ws;  ws += N2;                       //     3,000
  float* alpha = ws;  ws += LAYERS;

  const dim3 tile_grid(NGX, MT);

  hub_copy<<<(N1 * RANK + 255) / 256, 256, 0, stream>>>(U0, Ucur, N1 * RANK);
  hub_copy<<<(RANK * N2 + 255) / 256, 256, 0, stream>>>(V0, Vcur, RANK * N2);
  hub_alpha<<<1, 32, 0, stream>>>(c_list, alpha);
  hub_nobs<<<(N2 + 255) / 256, 256, 0, stream>>>(X, nobs);

  for (int layer = 0; layer < LAYERS; layer++) {
    // ---------------- column pass (update V) ----------------
    hub_gram_col<<<N2, 512, 0, stream>>>(X, Ucur, G);
    hub_invert<<<N2, 64, 0, stream>>>(G, Gi);
    hub_fill<<<(N2 + 255) / 256, 256, 0, stream>>>(sig, N2, sigma0);
    for (int it = 0; it < ITERS; it++) {
      hub_fill<<<(N2 + 255) / 256, 256, 0, stream>>>(ssum, N2, nullptr);
      hub_uv_psi<0><<<tile_grid, 32, 0, stream>>>(X, Ucur, Vcur, sig, c_list, layer,
                                                  Psi, ssum, nullptr);
      hub_sig<<<(N2 + 255) / 256, 256, 0, stream>>>(ssum, nobs, alpha, lamda,
                                                    layer, sig, N2, 0);
      hub_fill<<<(RANK * N2 + 255) / 256, 256, 0, stream>>>(T, RANK * N2, nullptr);
      hub_tcol<<<dim3((N2 + TC_COLS - 1) / TC_COLS, N1 / TC_ROWCHUNK), 320, 0, stream>>>(
          Psi, Ucur, T);
      hub_vupdate<<<(RANK * N2 + 255) / 256, 256, 0, stream>>>(Vcur, Gi, T, sig, mu, layer);
    }
    // ---------------- row pass (update U) ----------------
    hub_gram_row<<<N1, 512, 0, stream>>>(X, Vcur, G);
    hub_invert<<<N1, 64, 0, stream>>>(G, Gi);
    hub_fill<<<(N1 + 255) / 256, 256, 0, stream>>>(sig, N1, sigma0);
    for (int it = 0; it < ITERS; it++) {
      hub_fill<<<(N1 + 255) / 256, 256, 0, stream>>>(ssum, N1, nullptr);
      hub_uv_psi<1><<<tile_grid, 32, 0, stream>>>(X, Ucur, Vcur, sig, c_list, layer,
                                                  Psi, ssum, nullptr);
      hub_sig<<<(N1 + 255) / 256, 256, 0, stream>>>(ssum, nobs, alpha, lamda,
                                                    layer, sig, N1, 1);
      hub_trow<<<N1, 256, 0, stream>>>(Psi, Vcur, T);
      hub_uupdate<<<(N1 * RANK + 255) / 256, 256, 0, stream>>>(Ucur, Gi, T, sig, mu, layer);
    }
  }

  // final output: U @ V
  hub_uv_psi<2><<<tile_grid, 32, 0, stream>>>(X, Ucur, Vcur, sig, c_list, 0,
                                              Psi, ssum, (float*)d_out);
}